// RobustNormalEstimator_86431921865228
// MI455X (gfx1250) — compile-verified
//
#include <hip/hip_runtime.h>
#include <math.h>

#define BATCH 4
#define NPTS  8192
#define KNN   16
#define KK    17   // K+1 including self

typedef __attribute__((ext_vector_type(2))) float v2f;
typedef __attribute__((ext_vector_type(8))) float v8f;

// ---------------------------------------------------------------------------
// Kernel A: repack points as float4 (x, y, z, |p|^2) for aligned b128 loads
// ---------------------------------------------------------------------------
__global__ void pack_kernel(const float* __restrict__ pts, float4* __restrict__ pts4) {
  int t = blockIdx.x * blockDim.x + threadIdx.x;
  if (t < BATCH * NPTS) {
    float x = pts[t * 3 + 0], y = pts[t * 3 + 1], z = pts[t * 3 + 2];
    pts4[t] = make_float4(x, y, z, x * x + y * y + z * z);
  }
}

// ---------------------------------------------------------------------------
// Kernel B: KNN via WMMA Gram tiles + per-lane top-17 insertion lists.
// One wave32 per 16 query rows. For each 16-column tile:
//   G = A(16x4) * B(4x16) via V_WMMA_F32_16X16X4_F32  (K padded 3->4 with 0)
//   d2 = sq_i + sq_j - 2*G  staged through LDS,
//   lane L keeps top-17 of row (L&15), column-half (L>>4); merge at end.
// Fragment layout per ISA 7.12.2 (32-bit A 16x4): lanes 0-15 hold K=0,1;
// lanes 16-31 hold K=2,3. B (4x16) mirrored. C/D: vgpr r = row r+8*(lane>=16),
// col = lane&15. B-fragment build is branchless: one b128 load + cndmask.
// ---------------------------------------------------------------------------
__global__ __launch_bounds__(32) void knn_kernel(const float4* __restrict__ pts4,
                                                 int* __restrict__ knn_idx) {
  __shared__ float tile[256];
  __shared__ float md[32][KK];
  __shared__ int   mi[32][KK];

  const int lane = threadIdx.x;
  const int m    = lane & 15;
  const int hi   = lane >> 4;
  const int blk  = blockIdx.x;
  const int b    = blk / (NPTS / 16);
  const int i0   = (blk % (NPTS / 16)) * 16;

  const float4* p4 = pts4 + (size_t)b * NPTS;

  // A fragment: rows i0..i0+15, K = {x,y,z,0}
  v2f afrag;
  {
    float4 pa = p4[i0 + m];
    afrag.x = hi ? pa.z : pa.x;
    afrag.y = hi ? 0.0f : pa.y;
  }
  // row squared-norms this lane contributes d2 for (rows r + 8*hi)
  float srow[8];
#pragma unroll
  for (int r = 0; r < 8; ++r) srow[r] = p4[i0 + r + 8 * hi].w;

  // per-lane sorted top-KK list (ascending)
  float dlist[KK];
  int   ilist[KK];
#pragma unroll
  for (int t = 0; t < KK; ++t) { dlist[t] = 3.0e38f; ilist[t] = 0; }

  for (int j0 = 0; j0 < NPTS; j0 += 16) {
    // B fragment: candidate columns j0..j0+15 (branchless, single b128 load)
    const int col = j0 + m;
    float4 pb = p4[col];
    v2f bfrag;
    bfrag.x = hi ? pb.z : pb.x;
    bfrag.y = hi ? 0.0f : pb.y;
    const float sqj = pb.w;

    v8f c = {};
    c = __builtin_amdgcn_wmma_f32_16x16x4_f32(false, afrag, false, bfrag,
                                              (short)0, c, false, false);

#pragma unroll
    for (int r = 0; r < 8; ++r) {
      const int rowr = r + 8 * hi;
      float d2 = srow[r] + sqj - 2.0f * c[r];
      tile[rowr * 16 + m] = d2;
    }
    __syncthreads();

    // lane owns query row m, column half hi
#pragma unroll
    for (int t = 0; t < 8; ++t) {
      const int colh = hi * 8 + t;
      const float d = tile[m * 16 + colh];
      const int   j = j0 + colh;
      if (d < dlist[KK - 1]) {
        dlist[KK - 1] = d; ilist[KK - 1] = j;
#pragma unroll
        for (int s = KK - 1; s > 0; --s) {
          if (dlist[s] < dlist[s - 1]) {
            float td = dlist[s]; dlist[s] = dlist[s - 1]; dlist[s - 1] = td;
            int   ti = ilist[s]; ilist[s] = ilist[s - 1]; ilist[s - 1] = ti;
          }
        }
      }
    }
    __syncthreads();
  }

  // dump both half-lists, merge per row, drop nearest (self)
#pragma unroll
  for (int t = 0; t < KK; ++t) { md[lane][t] = dlist[t]; mi[lane][t] = ilist[t]; }
  __syncthreads();

  if (lane < 16) {
    int ia = 0, ib = 0;
    const int outbase = ((b * NPTS) + i0 + lane) * KNN;
    for (int t = 0; t < KK; ++t) {
      float da = (ia < KK) ? md[lane][ia]      : 3.0e38f;
      float db = (ib < KK) ? md[lane + 16][ib] : 3.0e38f;
      int j;
      if (da <= db) { j = mi[lane][ia];      ia++; }
      else          { j = mi[lane + 16][ib]; ib++; }
      if (t > 0) knn_idx[outbase + (t - 1)] = j;
    }
  }
}

// ---------------------------------------------------------------------------
// smallest-eigenvalue eigenvector of symmetric 3x3 (trig eigenvalues + cross)
// ---------------------------------------------------------------------------
__device__ inline void smallest_eigenvec(float a00, float a01, float a02,
                                         float a11, float a12, float a22,
                                         float& nx, float& ny, float& nz) {
  float p1 = a01 * a01 + a02 * a02 + a12 * a12;
  float q  = (a00 + a11 + a22) * (1.0f / 3.0f);
  float b00 = a00 - q, b11 = a11 - q, b22 = a22 - q;
  float p2 = b00 * b00 + b11 * b11 + b22 * b22 + 2.0f * p1;
  float p  = sqrtf(p2 * (1.0f / 6.0f));
  float lmin;
  if (p < 1e-20f) {
    lmin = q;
  } else {
    float ip = 1.0f / p;
    float c00 = b00 * ip, c01 = a01 * ip, c02 = a02 * ip;
    float c11 = b11 * ip, c12 = a12 * ip, c22 = b22 * ip;
    float detB = c00 * (c11 * c22 - c12 * c12)
               - c01 * (c01 * c22 - c12 * c02)
               + c02 * (c01 * c12 - c11 * c02);
    float r = 0.5f * detB;
    r = fminf(1.0f, fmaxf(-1.0f, r));
    float phi = acosf(r) * (1.0f / 3.0f);
    lmin = q + 2.0f * p * cosf(phi + 2.0943951023931953f);  // smallest
  }
  // rows of (A - lmin I), eigenvector = best cross product of row pairs
  float r0x = a00 - lmin, r0y = a01,        r0z = a02;
  float r1x = a01,        r1y = a11 - lmin, r1z = a12;
  float r2x = a02,        r2y = a12,        r2z = a22 - lmin;

  float c0x = r0y * r1z - r0z * r1y, c0y = r0z * r1x - r0x * r1z, c0z = r0x * r1y - r0y * r1x;
  float c1x = r0y * r2z - r0z * r2y, c1y = r0z * r2x - r0x * r2z, c1z = r0x * r2y - r0y * r2x;
  float c2x = r1y * r2z - r1z * r2y, c2y = r1z * r2x - r1x * r2z, c2z = r1x * r2y - r1y * r2x;

  float n0 = c0x * c0x + c0y * c0y + c0z * c0z;
  float n1 = c1x * c1x + c1y * c1y + c1z * c1z;
  float n2 = c2x * c2x + c2y * c2y + c2z * c2z;

  float vx = c0x, vy = c0y, vz = c0z, vn = n0;
  if (n1 > vn) { vx = c1x; vy = c1y; vz = c1z; vn = n1; }
  if (n2 > vn) { vx = c2x; vy = c2y; vz = c2z; vn = n2; }
  if (vn < 1e-30f) { nx = 0.0f; ny = 0.0f; nz = 1.0f; return; }
  float inv = rsqrtf(vn);
  nx = vx * inv; ny = vy * inv; nz = vz * inv;
}

// ---------------------------------------------------------------------------
// Kernel C: per-point MLP weighting + weighted 3x3 PCA normal (unsigned)
// ---------------------------------------------------------------------------
__global__ __launch_bounds__(256) void mlp_pca_kernel(
    const float4* __restrict__ pts4, const int* __restrict__ knn_idx,
    const float* __restrict__ W1, const float* __restrict__ B1,
    const float* __restrict__ W2, const float* __restrict__ B2,
    float* __restrict__ ntmp) {
  __shared__ float sW1[96], sB1[32], sW2[32], sB2[1];
  const int tid = threadIdx.x;
  if (tid < 96) sW1[tid] = W1[tid];
  if (tid < 32) { sB1[tid] = B1[tid]; sW2[tid] = W2[tid]; }
  if (tid == 0) sB2[0] = B2[0];
  __syncthreads();

  const int t = blockIdx.x * blockDim.x + tid;
  if (t >= BATCH * NPTS) return;
  const int b = t / NPTS;
  const int n = t - b * NPTS;
  const float4* p4 = pts4 + (size_t)b * NPTS;

  const float4 pc = p4[n];
  const int* idx = knn_idx + (size_t)t * KNN;

  float dx[KNN], dy[KNN], dz[KNN], acc[KNN];
  const float bb2 = sB2[0];
#pragma unroll
  for (int k = 0; k < KNN; ++k) {
    const float4 pj = p4[idx[k]];
    dx[k] = pj.x - pc.x;
    dy[k] = pj.y - pc.y;
    dz[k] = pj.z - pc.z;
    acc[k] = bb2;
  }
  // h = relu(diff @ W1 + b1); acc = h @ W2 + b2   (h-outer, k-inner)
  for (int h = 0; h < 32; ++h) {
    const float wx = sW1[h], wy = sW1[32 + h], wz = sW1[64 + h];
    const float bh = sB1[h], w2h = sW2[h];
#pragma unroll
    for (int k = 0; k < KNN; ++k) {
      float v = fmaf(dz[k], wz, fmaf(dy[k], wy, fmaf(dx[k], wx, bh)));
      acc[k] = fmaf(fmaxf(v, 0.0f), w2h, acc[k]);
    }
  }
  // cov = sum_k w^2 * d d^T / (K-1)
  float a00 = 0, a01 = 0, a02 = 0, a11 = 0, a12 = 0, a22 = 0;
#pragma unroll
  for (int k = 0; k < KNN; ++k) {
    const float sg = 1.0f / (1.0f + __expf(-acc[k]));
    const float w2 = sg * sg;
    a00 = fmaf(w2 * dx[k], dx[k], a00);
    a01 = fmaf(w2 * dx[k], dy[k], a01);
    a02 = fmaf(w2 * dx[k], dz[k], a02);
    a11 = fmaf(w2 * dy[k], dy[k], a11);
    a12 = fmaf(w2 * dy[k], dz[k], a12);
    a22 = fmaf(w2 * dz[k], dz[k], a22);
  }
  const float inv15 = 1.0f / (float)(KNN - 1);
  a00 *= inv15; a01 *= inv15; a02 *= inv15;
  a11 *= inv15; a12 *= inv15; a22 *= inv15;

  float nx, ny, nz;
  smallest_eigenvec(a00, a01, a02, a11, a12, a22, nx, ny, nz);
  ntmp[(size_t)t * 3 + 0] = nx;
  ntmp[(size_t)t * 3 + 1] = ny;
  ntmp[(size_t)t * 3 + 2] = nz;
}

// ---------------------------------------------------------------------------
// Kernel D: orient normals against batch reference normal (point 0)
// sign semantics match jnp.sign (sign(0) == 0)
// ---------------------------------------------------------------------------
__global__ void signfix_kernel(const float* __restrict__ ntmp, float* __restrict__ out) {
  const int t = blockIdx.x * blockDim.x + threadIdx.x;
  if (t >= BATCH * NPTS) return;
  const int b = t / NPTS;
  const size_t rb = (size_t)b * NPTS * 3;
  const float rx = ntmp[rb + 0], ry = ntmp[rb + 1], rz = ntmp[rb + 2];
  const float nx = ntmp[(size_t)t * 3 + 0];
  const float ny = ntmp[(size_t)t * 3 + 1];
  const float nz = ntmp[(size_t)t * 3 + 2];
  const float dot = nx * rx + ny * ry + nz * rz;
  const float s = (dot > 0.0f) ? 1.0f : ((dot < 0.0f) ? -1.0f : 0.0f);
  out[(size_t)t * 3 + 0] = nx * s;
  out[(size_t)t * 3 + 1] = ny * s;
  out[(size_t)t * 3 + 2] = nz * s;
}

// ---------------------------------------------------------------------------
extern "C" void kernel_launch(void* const* d_in, const int* in_sizes, int n_in,
                              void* d_out, int out_size, void* d_ws, size_t ws_size,
                              hipStream_t stream) {
  const float* points = (const float*)d_in[0];
  const float* W1 = (const float*)d_in[1];
  const float* B1 = (const float*)d_in[2];
  const float* W2 = (const float*)d_in[3];
  const float* B2 = (const float*)d_in[4];
  float* out = (float*)d_out;

  char* ws = (char*)d_ws;
  float4* pts4 = (float4*)ws;                                    // B*N*16 B (512 KB)
  int*    knn  = (int*)(ws + (size_t)BATCH * NPTS * 16);         // B*N*K*4 B (2 MB)
  float*  ntm  = (float*)(ws + (size_t)BATCH * NPTS * 16
                             + (size_t)BATCH * NPTS * KNN * 4);  // B*N*3*4 B (384 KB)

  const int total = BATCH * NPTS;
  pack_kernel<<<(total + 255) / 256, 256, 0, stream>>>(points, pts4);
  knn_kernel<<<total / 16, 32, 0, stream>>>(pts4, knn);
  mlp_pca_kernel<<<(total + 255) / 256, 256, 0, stream>>>(pts4, knn, W1, B1, W2, B2, ntm);
  signfix_kernel<<<(total + 255) / 256, 256, 0, stream>>>(ntm, out);
}